// UI_layer_50311246905588
// MI455X (gfx1250) — compile-verified
//
#include <hip/hip_runtime.h>

#define DVEC 128
#define N_USERS_C 100000

typedef float v2f __attribute__((ext_vector_type(2)));
typedef float v8f __attribute__((ext_vector_type(8)));

// ---------------------------------------------------------------------------
// Zero-fill the output (atomics accumulate into it).
// ---------------------------------------------------------------------------
__global__ __launch_bounds__(256) void zero_f32_kernel(float* __restrict__ out,
                                                       long long n) {
  long long i = (long long)blockIdx.x * blockDim.x + threadIdx.x;
  long long stride = (long long)gridDim.x * blockDim.x;
  for (; i < n; i += stride) out[i] = 0.0f;
}

// ---------------------------------------------------------------------------
// Main SpMM: each wave handles 16 nonzeros per iteration.
//   D(16x16 chunk) = diag(vals[16]) x G(16x16 chunk)  via 4x V_WMMA_F32_16X16X4_F32
// then coalesced global_atomic_add_f32 scatter of the 16 output rows.
// A 16x4 f32 layout: lane m=lane&15 = M; VGPR0 -> K=2*hi, VGPR1 -> K=2*hi+1.
// B  4x16 f32 layout: lane n=lane&15 = N; VGPR0 -> K=2*hi, VGPR1 -> K=2*hi+1.
// D 16x16 f32 layout: VGPR r -> (M = r + 8*hi, N = lane&15).
//
// Global accesses use 32-bit BYTE offsets against uniform SGPR bases
// (GVS addressing: SGPR_base + VGPR_I32 + IOFFSET). The per-chunk column
// offset is the compile-time constant c*64 bytes, so it can fold into the
// VMEM instruction's 24-bit immediate: one offset VGPR per stream reused
// across all 8 chunks, no per-chunk address VALU. Row/col bases are
// pre-scaled to bytes BEFORE the bpermute so post-shuffle math is one add.
// ---------------------------------------------------------------------------
__global__ __launch_bounds__(256) void spmm_wmma_kernel(
    const float* __restrict__ dense, const int* __restrict__ rows,
    const int* __restrict__ cols, const float* __restrict__ vals,
    float* __restrict__ out, int ngroups) {
  const int lane    = threadIdx.x & 31;
  const unsigned m  = lane & 15;   // M (for A) / N (for B,D)
  const int hi      = lane >> 4;   // half-wave selector
  const unsigned mB = m * 4u;      // byte offset of column m
  const int wave   = blockIdx.x * (blockDim.x >> 5) + (threadIdx.x >> 5);
  const int nwaves = gridDim.x * (blockDim.x >> 5);

  const char* denseB = (const char*)dense;
  char*       outB   = (char*)out;

  for (int g = wave; g < ngroups; g += nwaves) {
    const int i0 = g << 4;
    // Per-lane nonzero data (lanes 16..31 mirror lanes 0..15 -> shuffle-free A).
    const unsigned colByte = (unsigned)cols[i0 + (int)m] * (DVEC * 4u);
    const unsigned rowByte = (unsigned)rows[i0 + (int)m] * (DVEC * 4u);
    const float    val_l   = vals[i0 + (int)m];

    // A = diag(vals) split into 4 slices of 16x4 (K-global = 4*k0 + k_local).
    v2f a[4];
#pragma unroll
    for (int k0 = 0; k0 < 4; ++k0) {
      const unsigned kg = 4 * k0 + 2 * hi;
      a[k0].x = (m == kg)     ? val_l : 0.0f;
      a[k0].y = (m == kg + 1) ? val_l : 0.0f;
    }

    // 32-bit byte gather offsets: slice k0 reads rows cols[i0+4*k0+2*hi (+1)].
    unsigned offA[4], offB[4];
#pragma unroll
    for (int k0 = 0; k0 < 4; ++k0) {
      const int kg = 4 * k0 + 2 * hi;
      offA[k0] = (unsigned)__shfl((int)colByte, kg, 32) + mB;
      offB[k0] = (unsigned)__shfl((int)colByte, kg + 1, 32) + mB;
    }

    // 32-bit byte scatter offsets: D VGPR r -> output row rows[i0 + r + 8*hi].
    unsigned offD[8];
#pragma unroll
    for (int r = 0; r < 8; ++r) {
      offD[r] = (unsigned)__shfl((int)rowByte, r + 8 * hi, 32) + mB;
    }

#pragma unroll
    for (int c = 0; c < 8; ++c) {  // 8 chunks of 16 columns = D=128
      v8f acc = {};
#pragma unroll
      for (int k0 = 0; k0 < 4; ++k0) {
        v2f b;
        b.x = *(const float*)(denseB + offA[k0] + c * 64);
        b.y = *(const float*)(denseB + offB[k0] + c * 64);
        acc = __builtin_amdgcn_wmma_f32_16x16x4_f32(
            false, a[k0], false, b, (short)0, acc, false, false);
      }
      // Coalesced atomic scatter: VGPR r, lanes 0..15 cover 64 contiguous
      // bytes of one output row -> near line-rate RMW in L2.
#pragma unroll
      for (int r = 0; r < 8; ++r) {
        atomicAdd((float*)(outB + offD[r] + c * 64), acc[r]);
      }
    }
  }
}

// ---------------------------------------------------------------------------
// Tail: one wave per leftover nonzero (nnz % 16). b128 gather + 4 atomics/lane.
// ---------------------------------------------------------------------------
__global__ __launch_bounds__(256) void spmm_tail_kernel(
    const float* __restrict__ dense, const int* __restrict__ rows,
    const int* __restrict__ cols, const float* __restrict__ vals,
    float* __restrict__ out, int start, int nnz) {
  const int lane = threadIdx.x & 31;
  const int wave = blockIdx.x * (blockDim.x >> 5) + (threadIdx.x >> 5);
  const int i = start + wave;
  if (i >= nnz) return;
  const unsigned col = (unsigned)cols[i];
  const unsigned row = (unsigned)rows[i];
  const float    v   = vals[i];
  const float4* src = (const float4*)(dense + (size_t)col * DVEC);
  const float4 gth = src[lane];
  const unsigned dst = row * DVEC + (unsigned)lane * 4;
  atomicAdd(out + dst + 0, gth.x * v);
  atomicAdd(out + dst + 1, gth.y * v);
  atomicAdd(out + dst + 2, gth.z * v);
  atomicAdd(out + dst + 3, gth.w * v);
}

// ---------------------------------------------------------------------------
extern "C" void kernel_launch(void* const* d_in, const int* in_sizes, int n_in,
                              void* d_out, int out_size, void* d_ws,
                              size_t ws_size, hipStream_t stream) {
  const float* input  = (const float*)d_in[0];
  const int*   u_rows = (const int*)d_in[1];
  const int*   u_cols = (const int*)d_in[2];
  const float* u_vals = (const float*)d_in[3];
  const int*   i_rows = (const int*)d_in[4];
  const int*   i_cols = (const int*)d_in[5];
  const float* i_vals = (const float*)d_in[6];
  float*       out    = (float*)d_out;

  const int nnz_u = in_sizes[1];
  const int nnz_i = in_sizes[4];
  const long long total = (long long)out_size;

  // 1) zero the accumulation target
  {
    int blocks = (int)((total + 255) / 256);
    if (blocks > 8192) blocks = 8192;
    zero_f32_kernel<<<blocks, 256, 0, stream>>>(out, total);
  }

  const float* users_emb = input;                                // rows [0, N_USERS)
  const float* items_emb = input + (long long)N_USERS_C * DVEC;  // rows [N_USERS, ...)
  float* out_u = out;
  float* out_i = out + (long long)N_USERS_C * DVEC;

  const int blocks_main = 4096;  // 32768 wave32 -> fills the device, grid-stride

  // 2) users SpMM
  {
    const int groups = nnz_u >> 4, rem = nnz_u & 15;
    if (groups > 0)
      spmm_wmma_kernel<<<blocks_main, 256, 0, stream>>>(users_emb, u_rows,
                                                        u_cols, u_vals, out_u,
                                                        groups);
    if (rem > 0)
      spmm_tail_kernel<<<(rem + 7) / 8, 256, 0, stream>>>(
          users_emb, u_rows, u_cols, u_vals, out_u, groups << 4, nnz_u);
  }

  // 3) items SpMM
  {
    const int groups = nnz_i >> 4, rem = nnz_i & 15;
    if (groups > 0)
      spmm_wmma_kernel<<<blocks_main, 256, 0, stream>>>(items_emb, i_rows,
                                                        i_cols, i_vals, out_i,
                                                        groups);
    if (rem > 0)
      spmm_tail_kernel<<<(rem + 7) / 8, 256, 0, stream>>>(
          items_emb, i_rows, i_cols, i_vals, out_i, groups << 4, nnz_i);
  }
}